// ALSTM_16887811408328
// MI455X (gfx1250) — compile-verified
//
#include <hip/hip_runtime.h>
#include <cstddef>
#include <cstdint>

typedef __attribute__((ext_vector_type(2))) float v2f;
typedef __attribute__((ext_vector_type(8))) float v8f;

#define B_DIM   4096
#define H_DIM   512
#define G_DIM   2048   // 4*H
#define NSTEPS  16
#define TS      20     // padded LDS tile stride (floats): 16B-aligned rows, conflict-free

__device__ __forceinline__ float sigf(float x) { return 1.0f / (1.0f + __expf(-x)); }

// low 32 bits of a flat shared-memory address == LDS byte offset (HW truncates)
__device__ __forceinline__ uint32_t lds_off32(const void* p) {
  return (uint32_t)(uintptr_t)p;
}

// CDNA5 async global->LDS (ASYNCcnt-tracked, no VGPR staging).  GVS mode:
// mem_addr = SADDR + VADDR(u32 byte offset);  VDST vgpr = LDS byte address.
__device__ __forceinline__ void async_b128(uint32_t lds, uint64_t sbase, uint32_t voff) {
  asm volatile("global_load_async_to_lds_b128 %0, %1, %2"
               :: "v"(lds), "v"(voff), "s"(sbase) : "memory");
}
__device__ __forceinline__ void async_b32(uint32_t lds, uint64_t sbase, uint32_t voff) {
  asm volatile("global_load_async_to_lds_b32 %0, %1, %2"
               :: "v"(lds), "v"(voff), "s"(sbase) : "memory");
}

// ---------------------------------------------------------------------------
// Generic NT GEMM on the fp32 WMMA path: out[m][n] = A[m][:]·W[n][:] + bias[n]
// 128x128 block tile, K slab 16, double-buffered LDS filled by async loads.
// 8 wave32s; each wave = 64x32 patch = 4x2 grid of V_WMMA_F32_16X16X4_F32.
// ALW: W rows 16B-aligned (ldw % 4 == 0) -> b128 fills, else b32 fills.
// ---------------------------------------------------------------------------
template<bool ALW>
__global__ __launch_bounds__(256)
void gemm_nt_f32(const float* __restrict__ A, const float* __restrict__ W, int ldw,
                 const float* __restrict__ bias, float* __restrict__ out, int N, int K)
{
  __shared__ float smem[4 * 128 * TS];
  auto la = [&](int b) -> float* { return smem + b * (128 * TS); };
  auto lb = [&](int b) -> float* { return smem + (2 + b) * (128 * TS); };

  const int tid  = threadIdx.x;
  const int lane = tid & 31, wave = tid >> 5;
  const int wm   = wave >> 2, wn = wave & 3;
  const int l    = lane & 15, half = lane >> 4;
  const int m0   = blockIdx.y * 128;
  const int n0   = blockIdx.x * 128;
  const int ar   = tid >> 2, aq = tid & 3;     // item: row ar(+64), float4 quad aq

  const uint64_t Ab = (uint64_t)A;
  const uint64_t Wb = (uint64_t)W;

  v8f acc[4][2];
  #pragma unroll
  for (int mt = 0; mt < 4; ++mt)
    #pragma unroll
    for (int nt = 0; nt < 2; ++nt)
      acc[mt][nt] = (v8f){0.f,0.f,0.f,0.f,0.f,0.f,0.f,0.f};

  auto issue = [&](int kb, int buf) {
    float* ab = la(buf);
    float* bb = lb(buf);
    #pragma unroll
    for (int s = 0; s < 2; ++s) {
      int r = ar + s * 64;
      async_b128(lds_off32(ab + r * TS + aq * 4), Ab,
                 (uint32_t)(((m0 + r) * K + kb + aq * 4) * 4));
    }
    if constexpr (ALW) {
      #pragma unroll
      for (int s = 0; s < 2; ++s) {
        int r = ar + s * 64;
        async_b128(lds_off32(bb + r * TS + aq * 4), Wb,
                   (uint32_t)(((n0 + r) * ldw + kb + aq * 4) * 4));
      }
    } else {
      #pragma unroll
      for (int s = 0; s < 8; ++s) {
        int e = tid + s * 256;
        int r = e >> 4, cc = e & 15;
        async_b32(lds_off32(bb + r * TS + cc), Wb,
                  (uint32_t)(((n0 + r) * ldw + kb + cc) * 4));
      }
    }
  };

  auto compute = [&](const float* abuf, const float* bbuf) {
    #pragma unroll
    for (int ks = 0; ks < 16; ks += 4) {
      v2f bf[2];
      #pragma unroll
      for (int nt = 0; nt < 2; ++nt)
        bf[nt] = *(const v2f*)&bbuf[(wn * 32 + nt * 16 + l) * TS + ks + 2 * half];
      #pragma unroll
      for (int mt = 0; mt < 4; ++mt) {
        v2f af = *(const v2f*)&abuf[(wm * 64 + mt * 16 + l) * TS + ks + 2 * half];
        #pragma unroll
        for (int nt = 0; nt < 2; ++nt)
          acc[mt][nt] = __builtin_amdgcn_wmma_f32_16x16x4_f32(
              false, af, false, bf[nt], (short)0, acc[mt][nt], false, false);
      }
    }
  };

  const int nslab = K >> 4;
  issue(0, 0);
  for (int i = 0; i < nslab; ++i) {
    if (i + 1 < nslab) {
      issue((i + 1) << 4, (i + 1) & 1);
      if constexpr (ALW) asm volatile("s_wait_asynccnt 4"  ::: "memory");
      else               asm volatile("s_wait_asynccnt 10" ::: "memory");
    } else {
      asm volatile("s_wait_asynccnt 0" ::: "memory");
    }
    __syncthreads();                       // slab i visible to all waves
    compute(la(i & 1), lb(i & 1));
    __syncthreads();                       // buffer reusable
  }

  #pragma unroll
  for (int mt = 0; mt < 4; ++mt)
    #pragma unroll
    for (int nt = 0; nt < 2; ++nt) {
      int col  = n0 + wn * 32 + nt * 16 + l;
      int rowb = m0 + wm * 64 + mt * 16 + half * 8;
      float bv = bias[col];
      #pragma unroll
      for (int r = 0; r < 8; ++r)
        out[(size_t)(rowb + r) * N + col] = acc[mt][nt][r] + bv;
    }
}

// ---------------------------------------------------------------------------
// Fused recurrent step: for a 128-row x 32-col slice of H, compute all four
// gate columns {g*512+j}, add bias + x_part, stage the 128x128 gate patch in
// LDS, then apply the LSTM cell update:
//   c = sig(f)*c + sig(i)*tanh(g);  h_out = sig(o)*tanh(c)
// h is ping-pong buffered (hin read by all blocks, hout written per block).
// ---------------------------------------------------------------------------
__global__ __launch_bounds__(256)
void gates_lstm_fused(const float* __restrict__ hin, const float* __restrict__ Whh,
                      const float* __restrict__ bias, const float* __restrict__ xpart,
                      float* __restrict__ c, float* __restrict__ hout)
{
  __shared__ float smem[128 * 132];        // >= 4*128*TS (10240); gate patch 16896
  auto la = [&](int b) -> float* { return smem + b * (128 * TS); };
  auto lb = [&](int b) -> float* { return smem + (2 + b) * (128 * TS); };

  const int tid  = threadIdx.x;
  const int lane = tid & 31, wave = tid >> 5;
  const int wm   = wave >> 2, wn = wave & 3;   // wn == gate index g
  const int l    = lane & 15, half = lane >> 4;
  const int m0   = blockIdx.y * 128;
  const int n0   = blockIdx.x * 32;            // column chunk within H
  const int ar   = tid >> 2, aq = tid & 3;

  const uint64_t Ab = (uint64_t)hin;
  const uint64_t Wb = (uint64_t)Whh;

  v8f acc[4][2];
  #pragma unroll
  for (int mt = 0; mt < 4; ++mt)
    #pragma unroll
    for (int nt = 0; nt < 2; ++nt)
      acc[mt][nt] = (v8f){0.f,0.f,0.f,0.f,0.f,0.f,0.f,0.f};

  auto issue = [&](int kb, int buf) {
    float* ab = la(buf);
    float* bb = lb(buf);
    #pragma unroll
    for (int s = 0; s < 2; ++s) {
      int r = ar + s * 64;
      async_b128(lds_off32(ab + r * TS + aq * 4), Ab,
                 (uint32_t)(((m0 + r) * H_DIM + kb + aq * 4) * 4));
      // W tile row r -> gate (r>>5), column chunk n0 + (r&31)
      int wrow = (r >> 5) * H_DIM + n0 + (r & 31);
      async_b128(lds_off32(bb + r * TS + aq * 4), Wb,
                 (uint32_t)((wrow * H_DIM + kb + aq * 4) * 4));
    }
  };

  auto compute = [&](const float* abuf, const float* bbuf) {
    #pragma unroll
    for (int ks = 0; ks < 16; ks += 4) {
      v2f bf[2];
      #pragma unroll
      for (int nt = 0; nt < 2; ++nt)
        bf[nt] = *(const v2f*)&bbuf[(wn * 32 + nt * 16 + l) * TS + ks + 2 * half];
      #pragma unroll
      for (int mt = 0; mt < 4; ++mt) {
        v2f af = *(const v2f*)&abuf[(wm * 64 + mt * 16 + l) * TS + ks + 2 * half];
        #pragma unroll
        for (int nt = 0; nt < 2; ++nt)
          acc[mt][nt] = __builtin_amdgcn_wmma_f32_16x16x4_f32(
              false, af, false, bf[nt], (short)0, acc[mt][nt], false, false);
      }
    }
  };

  const int nslab = H_DIM >> 4;
  issue(0, 0);
  for (int i = 0; i < nslab; ++i) {
    if (i + 1 < nslab) {
      issue((i + 1) << 4, (i + 1) & 1);
      asm volatile("s_wait_asynccnt 4" ::: "memory");
    } else {
      asm volatile("s_wait_asynccnt 0" ::: "memory");
    }
    __syncthreads();
    compute(la(i & 1), lb(i & 1));
    __syncthreads();
  }

  // phase 1: accumulators (+bias +x_part) -> LDS gate patch [128 rows][4*32 cols]
  float* const gt = smem;                  // stride 132: conflict-free
  #pragma unroll
  for (int mt = 0; mt < 4; ++mt)
    #pragma unroll
    for (int nt = 0; nt < 2; ++nt) {
      int lc   = nt * 16 + l;              // 0..31 within chunk
      int gcol = wn * H_DIM + n0 + lc;     // column in gates space [0,2048)
      float bv = bias[gcol];
      int rowb = wm * 64 + mt * 16 + half * 8;
      #pragma unroll
      for (int r = 0; r < 8; ++r) {
        int row = rowb + r;
        float v = acc[mt][nt][r] + bv + xpart[(size_t)(m0 + row) * G_DIM + gcol];
        gt[row * 132 + wn * 32 + lc] = v;
      }
    }
  __syncthreads();

  // phase 2: cell update, 16 cells per thread, coalesced global accesses
  #pragma unroll
  for (int s = 0; s < 16; ++s) {
    int id  = tid + s * 256;               // 0..4095
    int row = id >> 5, wc = id & 31;
    const float* g = &gt[row * 132 + wc];
    float ig = sigf(g[0]);
    float fg = sigf(g[32]);
    float gg = tanhf(g[64]);
    float og = sigf(g[96]);
    size_t ci = (size_t)(m0 + row) * H_DIM + n0 + wc;
    float cn = fg * c[ci] + ig * gg;
    c[ci]   = cn;
    hout[ci] = og * tanhf(cn);
  }
}

// ---------------------------------------------------------------------------
// Per-row halting + online blend accumulation. One wave32 per batch row.
// ---------------------------------------------------------------------------
__global__ __launch_bounds__(256)
void halt_accum(const float* __restrict__ h, const float* __restrict__ c,
                const float* __restrict__ Whalt, const float* __restrict__ bhalt,
                float* __restrict__ cum, float* __restrict__ psum, float* __restrict__ halted,
                float* __restrict__ hfin, float* __restrict__ cfin, float* __restrict__ ponder,
                int step, int last)
{
  const int wave = threadIdx.x >> 5;
  const int lane = threadIdx.x & 31;
  const int b    = blockIdx.x * 8 + wave;

  const float* hr = h + (size_t)b * H_DIM;
  float s = 0.f;
  #pragma unroll
  for (int j = lane; j < H_DIM; j += 32) s += hr[j] * Whalt[j];
  #pragma unroll
  for (int off = 16; off; off >>= 1) s += __shfl_xor(s, off, 32);

  float p = 0.f;
  if (lane == 0) {
    if (halted[b] == 0.f) {
      float hv = sigf(s + bhalt[0]);
      float cm = cum[b] + hv;
      cum[b] = cm;
      if (cm >= 1.0f - 0.01f || step == last) {
        p = 1.f - psum[b];                 // remainder r
        halted[b] = 1.f;
        ponder[b] = (float)step + 1.f + p;
      } else {
        p = hv;
        psum[b] += hv;
      }
    }
  }
  p = __shfl(p, 0, 32);

  const float* cr = c + (size_t)b * H_DIM;
  float* hf = hfin + (size_t)b * H_DIM;
  float* cf = cfin + (size_t)b * H_DIM;
  #pragma unroll
  for (int j = lane; j < H_DIM; j += 32) {
    hf[j] += p * hr[j];
    cf[j] += p * cr[j];
  }
}

// ---------------------------------------------------------------------------
__global__ __launch_bounds__(256)
void init_state(const float* __restrict__ h0, const float* __restrict__ c0,
                float* __restrict__ h, float* __restrict__ c,
                float* __restrict__ hfin, float* __restrict__ cfin,
                float* __restrict__ cum, float* __restrict__ psum, float* __restrict__ halted)
{
  size_t t = (size_t)blockIdx.x * blockDim.x + threadIdx.x;
  if (t < (size_t)B_DIM * H_DIM) {
    h[t] = h0[t]; c[t] = c0[t]; hfin[t] = 0.f; cfin[t] = 0.f;
  }
  if (t < B_DIM) { cum[t] = 0.f; psum[t] = 0.f; halted[t] = 0.f; }
}

__global__ __launch_bounds__(256)
void prep_bias0(const float* __restrict__ b_hh, const float* __restrict__ W_ih,
                float* __restrict__ b0)
{
  int n = blockIdx.x * blockDim.x + threadIdx.x;
  if (n < G_DIM) b0[n] = b_hh[n] + W_ih[(size_t)n * 513];   // w_flag = W_ih[:,0]
}

// ---------------------------------------------------------------------------
extern "C" void kernel_launch(void* const* d_in, const int* in_sizes, int n_in,
                              void* d_out, int out_size, void* d_ws, size_t ws_size,
                              hipStream_t stream)
{
  (void)in_sizes; (void)n_in; (void)out_size; (void)ws_size;

  const float* x      = (const float*)d_in[0];
  const float* h0     = (const float*)d_in[1];
  const float* c0     = (const float*)d_in[2];
  const float* W_ih   = (const float*)d_in[3];
  const float* b_ih   = (const float*)d_in[4];
  const float* W_hh   = (const float*)d_in[5];
  const float* b_hh   = (const float*)d_in[6];
  const float* W_halt = (const float*)d_in[7];
  const float* b_halt = (const float*)d_in[8];
  const float* W_out  = (const float*)d_in[9];
  const float* b_out  = (const float*)d_in[10];

  float* ws     = (float*)d_ws;
  float* xpart  = ws;                                       // B*4H
  float* hb0    = xpart + (size_t)B_DIM * G_DIM;            // B*H (ping)
  float* hb1    = hb0 + (size_t)B_DIM * H_DIM;              // B*H (pong)
  float* c      = hb1 + (size_t)B_DIM * H_DIM;              // B*H
  float* b0     = c + (size_t)B_DIM * H_DIM;                // 4H
  float* cum    = b0 + G_DIM;                               // B
  float* psum   = cum + B_DIM;                              // B
  float* halted = psum + B_DIM;                             // B

  float* outp   = (float*)d_out;                            // B*O
  float* hfin   = outp + (size_t)B_DIM * H_DIM;             // B*H (O==H)
  float* cfin   = hfin + (size_t)B_DIM * H_DIM;             // B*H
  float* ponder = cfin + (size_t)B_DIM * H_DIM;             // B

  init_state<<<(B_DIM * H_DIM) / 256, 256, 0, stream>>>(h0, c0, hb0, c, hfin, cfin,
                                                        cum, psum, halted);
  prep_bias0<<<(G_DIM + 255) / 256, 256, 0, stream>>>(b_hh, W_ih, b0);

  // x_part = x @ W_ih[:,1:].T + b_ih   (ldw = 513 -> unaligned W path)
  gemm_nt_f32<false><<<dim3(G_DIM / 128, B_DIM / 128), 256, 0, stream>>>(
      x, W_ih + 1, 513, b_ih, xpart, G_DIM, H_DIM);

  for (int n = 0; n < NSTEPS; ++n) {
    const float* hprev = (n & 1) ? hb1 : hb0;
    float*       hnext = (n & 1) ? hb0 : hb1;
    gates_lstm_fused<<<dim3(H_DIM / 32, B_DIM / 128), 256, 0, stream>>>(
        hprev, W_hh, (n == 0 ? b0 : b_hh), xpart, c, hnext);
    halt_accum<<<B_DIM / 8, 256, 0, stream>>>(hnext, c, W_halt, b_halt,
                                              cum, psum, halted,
                                              hfin, cfin, ponder, n, NSTEPS - 1);
  }

  // output = h_fin @ W_out.T + b_out  (sum_n p == 1 exactly)
  gemm_nt_f32<true><<<dim3(H_DIM / 128, B_DIM / 128), 256, 0, stream>>>(
      hfin, W_out, H_DIM, b_out, outp, H_DIM, H_DIM);
}